// CNNLayer_30296699306356
// MI455X (gfx1250) — compile-verified
//
#include <hip/hip_runtime.h>

typedef __attribute__((ext_vector_type(2))) float v2f;
typedef __attribute__((ext_vector_type(8))) float v8f;

#define CH        8
#define BATCH     16
#define NLVL      10
#define ACT_ELEMS (BATCH * 8192)   // 131072 floats = 512 KB per ping-pong buffer

// ---------------------------------------------------------------------------
// Kernel 1: input interpolation conv. [16,2048,1] -> [16,1024,8], relu.
// ---------------------------------------------------------------------------
__global__ __launch_bounds__(256) void input_conv_kernel(
    const float* __restrict__ in, const float* __restrict__ w,
    const float* __restrict__ bias, float* __restrict__ y) {
  int t = blockIdx.x * blockDim.x + threadIdx.x;   // 131072 threads
  int m = t >> 3;                                  // b*1024 + i
  int c = t & 7;
  int b = m >> 10;
  int i = m & 1023;
  float x0 = in[b * 2048 + 2 * i];
  float x1 = in[b * 2048 + 2 * i + 1];
  float acc = fmaf(x1, w[CH + c], fmaf(x0, w[c], bias[c]));
  y[t] = acc > 0.f ? acc : 0.f;
}

// ---------------------------------------------------------------------------
// Kernel 2: fused GEMM + bias + relu via V_WMMA_F32_16X16X4_F32, templated on
// the compile-time K (== N) of each butterfly level.
//
// Y[M,N] = relu(X[M,K] * W[K,N] + bias[N]).
// Block = 128 threads = 4 waves; one 16x16 C tile per block; the 4 waves each
// take a contiguous K/4 slice, accumulating into private C frags (two
// accumulator chains for ILP), reduced through LDS at the end.
//
// Frag layouts (ISA 7.12.2, wave32):
//   A 16x4 : lane<16 -> M=lane, K = k,k+1 ; lane>=16 -> M=lane-16, K = k+2,k+3
//   B 4x16 : VGPR0 = row k (lanes 0-15) / row k+2 (lanes 16-31); VGPR1 = k+1 / k+3
//   C/D    : VGPR r -> M = r + 8*(lane>=16), N = lane%16
// ---------------------------------------------------------------------------
template <int K>
__global__ __launch_bounds__(128) void gemm_bias_relu_wmma(
    const float* __restrict__ X, const float* __restrict__ W,
    const float* __restrict__ bias, float* __restrict__ Y, int M) {
  constexpr int N = K;
  constexpr int kPerWave = K / 4;
  __shared__ float red[4 * 256];

  const int lane = threadIdx.x & 31;
  const int wave = threadIdx.x >> 5;
  const int n0 = blockIdx.x * 16;
  const int m0 = blockIdx.y * 16;
  const int lr = lane & 15;       // 0..15
  const int hi = lane >> 4;       // 0 or 1 (lane half)
  const int kBeg = wave * kPerWave;

  // Per-lane base pointers; all subsequent offsets are compile-time scaled.
  const float* __restrict__ Arow = X + (size_t)(m0 + lr) * K + 2 * hi;   // A[m, kBeg+kk+2hi]
  const float* __restrict__ Bcol = W + (size_t)(2 * hi) * N + n0 + lr;   // W[kk+2hi, n]

  v8f acc0 = {};
  v8f acc1 = {};

  if constexpr (kPerWave >= 8) {
#pragma unroll 4
    for (int kk = 0; kk < kPerWave; kk += 8) {
      const int k0 = kBeg + kk;
      const int k1 = k0 + 4;
      // A: contiguous float2 per lane -> b64 loads
      v2f a0, a1;
      a0.x = Arow[k0];
      a0.y = Arow[k0 + 1];
      a1.x = Arow[k1];
      a1.y = Arow[k1 + 1];
      // B: two rows per frag, stride N (compile-time)
      v2f b0, b1;
      b0.x = Bcol[(size_t)k0 * N];
      b0.y = Bcol[(size_t)(k0 + 1) * N];
      b1.x = Bcol[(size_t)k1 * N];
      b1.y = Bcol[(size_t)(k1 + 1) * N];
      // one speculative prefetch per body (16 K-steps ahead)
      __builtin_prefetch(&Bcol[(size_t)(k0 + 16) * N], 0, 1);
      acc0 = __builtin_amdgcn_wmma_f32_16x16x4_f32(
          false, a0, false, b0, (short)0, acc0, false, false);
      acc1 = __builtin_amdgcn_wmma_f32_16x16x4_f32(
          false, a1, false, b1, (short)0, acc1, false, false);
    }
  } else {
    // kPerWave == 4 (level 1): single K-step per wave.
    v2f a0, b0;
    a0.x = Arow[kBeg];
    a0.y = Arow[kBeg + 1];
    b0.x = Bcol[(size_t)kBeg * N];
    b0.y = Bcol[(size_t)(kBeg + 1) * N];
    acc0 = __builtin_amdgcn_wmma_f32_16x16x4_f32(
        false, a0, false, b0, (short)0, acc0, false, false);
  }

  // Avoid dead fp adds in the single-step instantiation (x+0 is not an
  // identity the compiler may fold without fast-math).
  v8f acc;
  if constexpr (kPerWave >= 8) {
    acc = acc0 + acc1;
  } else {
    acc = acc0;
  }

  // Stash each wave's C frag, reduce the 4 partials, bias + relu, store.
#pragma unroll
  for (int r = 0; r < 8; ++r)
    red[wave * 256 + r * 32 + lane] = acc[r];
  __syncthreads();

  for (int s = threadIdx.x; s < 256; s += 128) {
    int vr = s >> 5;                 // C/D VGPR index 0..7
    int ln = s & 31;
    int mm = vr + 8 * (ln >> 4);
    int nn = ln & 15;
    float sum = red[s] + red[256 + s] + red[512 + s] + red[768 + s];
    sum += bias[n0 + nn];
    Y[(size_t)(m0 + mm) * N + n0 + nn] = sum > 0.f ? sum : 0.f;
  }
}

// ---------------------------------------------------------------------------
// Kernel 3: per-block dense heads. v[b,k,0:8] dot fea[k,0:8,o] -> out[b,k,o]
// ---------------------------------------------------------------------------
__global__ __launch_bounds__(256) void fea_dense_kernel(
    const float* __restrict__ X, const float* __restrict__ fea,
    float* __restrict__ out) {
  int t = blockIdx.x * blockDim.x + threadIdx.x;   // 32768 threads
  int b = t >> 11;
  int r = t & 2047;
  int k = r >> 1;
  int o = r & 1;
  const float* v = X + b * 8192 + k * CH;
  const float* f = fea + k * CH * 2 + o;
  float acc = 0.f;
#pragma unroll
  for (int c = 0; c < CH; ++c) acc = fmaf(v[c], f[2 * c], acc);
  out[t] = acc;
}

// ---------------------------------------------------------------------------
// Launcher. Input order: in_data, in_filter, in_bias, f1,b1,...,f10,b10, fea.
// ---------------------------------------------------------------------------
extern "C" void kernel_launch(void* const* d_in, const int* in_sizes, int n_in,
                              void* d_out, int out_size, void* d_ws, size_t ws_size,
                              hipStream_t stream) {
  (void)in_sizes; (void)n_in; (void)out_size; (void)ws_size;

  const float* in_data   = (const float*)d_in[0];
  const float* in_filter = (const float*)d_in[1];
  const float* in_bias   = (const float*)d_in[2];
  const float* fea       = (const float*)d_in[23];
  float* out = (float*)d_out;

  float* buf0 = (float*)d_ws;
  float* buf1 = buf0 + ACT_ELEMS;

  // Level 0: input conv -> buf0 holds [16,1024,8]
  input_conv_kernel<<<ACT_ELEMS / 256, 256, 0, stream>>>(
      in_data, in_filter, in_bias, buf0);

  // Butterfly levels 1..10: GEMM  M = 16384>>l, K = N = 8<<l
  float* src = buf0;
  float* dst = buf1;
  for (int l = 1; l <= NLVL; ++l) {
    const float* fw = (const float*)d_in[3 + 2 * (l - 1)];
    const float* fb = (const float*)d_in[4 + 2 * (l - 1)];
    int M = 16384 >> l;
    int KN = CH << l;                 // K == N
    dim3 grid(KN / 16, M / 16);       // always 512 blocks = 2048 waves
    switch (KN) {
      case 16:   gemm_bias_relu_wmma<16>  <<<grid, 128, 0, stream>>>(src, fw, fb, dst, M); break;
      case 32:   gemm_bias_relu_wmma<32>  <<<grid, 128, 0, stream>>>(src, fw, fb, dst, M); break;
      case 64:   gemm_bias_relu_wmma<64>  <<<grid, 128, 0, stream>>>(src, fw, fb, dst, M); break;
      case 128:  gemm_bias_relu_wmma<128> <<<grid, 128, 0, stream>>>(src, fw, fb, dst, M); break;
      case 256:  gemm_bias_relu_wmma<256> <<<grid, 128, 0, stream>>>(src, fw, fb, dst, M); break;
      case 512:  gemm_bias_relu_wmma<512> <<<grid, 128, 0, stream>>>(src, fw, fb, dst, M); break;
      case 1024: gemm_bias_relu_wmma<1024><<<grid, 128, 0, stream>>>(src, fw, fb, dst, M); break;
      case 2048: gemm_bias_relu_wmma<2048><<<grid, 128, 0, stream>>>(src, fw, fb, dst, M); break;
      case 4096: gemm_bias_relu_wmma<4096><<<grid, 128, 0, stream>>>(src, fw, fb, dst, M); break;
      case 8192: gemm_bias_relu_wmma<8192><<<grid, 128, 0, stream>>>(src, fw, fb, dst, M); break;
    }
    float* tmp = src; src = dst; dst = tmp;
  }

  // src now holds [16, 8192]; per-block heads -> out [16*2048]
  fea_dense_kernel<<<BATCH * 2048 / 256, 256, 0, stream>>>(src, fea, out);
}